// GraphTransfer_40664750359215
// MI455X (gfx1250) — compile-verified
//
#include <hip/hip_runtime.h>
#include <hip/hip_bf16.h>

typedef __attribute__((ext_vector_type(2))) float v2f;
typedef __attribute__((ext_vector_type(8))) float v8f;

#define NN 100000
#define EE 1600000
#define HD 128
#define NH (NN * HD)

// ---------------------------------------------------------------- utilities
__global__ void zero_f32(float* __restrict__ p, int n) {
    int i = blockIdx.x * blockDim.x + threadIdx.x;
    int stride = gridDim.x * blockDim.x;
    for (; i < n; i += stride) p[i] = 0.0f;
}

// ------------------------------------------------- edge scatter: agg[dst] += h[src]
// one thread handles one (edge, 4-channel) group: float4 load + 4 atomic f32 adds
__global__ void scatter_add(const float* __restrict__ h, const int* __restrict__ src,
                            const int* __restrict__ dst, float* __restrict__ agg) {
    int gid = blockIdx.x * blockDim.x + threadIdx.x;   // E*32 = 51.2M < 2^31
    int e  = gid >> 5;
    int c4 = (gid & 31) * 4;
    if (e >= EE) return;
    int s = src[e], d = dst[e];
    const float4 v = *(const float4*)(h + (long)s * HD + c4);
    float* a = agg + (long)d * HD + c4;
    atomicAdd(a + 0, v.x);
    atomicAdd(a + 1, v.y);
    atomicAdd(a + 2, v.z);
    atomicAdd(a + 3, v.w);
}

// ---------------------------------------------------------------- WMMA GEMM
// out[N,128] = concat(hin, agg)[N,kdim] @ W[kdim,128] + bias  (agg==null -> kdim=128)
// block = 256 threads = 8 waves; block owns a 16-row strip; wave w owns cols [16w,16w+16)
__global__ void __launch_bounds__(256)
gemm_mp(const float* __restrict__ hin, const float* __restrict__ agg,
        const float* __restrict__ W, const float* __restrict__ bias,
        float* __restrict__ out, int kdim, int do_relu) {
    __shared__ float Ap[16][260];      // pad stride 260: (4m+k)%64 -> conflict-free
    const int tid  = threadIdx.x;
    const int row0 = blockIdx.x * 16;

    // stage A panel (coalesced: consecutive threads -> consecutive channels)
    for (int i = tid; i < 16 * kdim; i += 256) {
        int r = i / kdim;
        int c = i - r * kdim;
        float v = (c < HD) ? hin[(long)(row0 + r) * HD + c]
                           : agg[(long)(row0 + r) * HD + (c - HD)];
        Ap[r][c] = v;
    }
    __syncthreads();

    const int wave = tid >> 5;
    const int lane = tid & 31;
    const int half = lane >> 4;     // 0: K pair {0,1} / M 0..7 ; 1: K pair {2,3} / M 8..15
    const int mn   = lane & 15;
    const int col0 = wave * 16;

    v8f c = {};
    for (int k0 = 0; k0 < kdim; k0 += 4) {
        int ka = k0 + half * 2;
        v2f a, b;
        a[0] = Ap[mn][ka];
        a[1] = Ap[mn][ka + 1];
        b[0] = W[ka * HD + col0 + mn];
        b[1] = W[(ka + 1) * HD + col0 + mn];
        // V_WMMA_F32_16X16X4_F32: (neg_a, A, neg_b, B, c_mod, C, reuse_a, reuse_b)
        c = __builtin_amdgcn_wmma_f32_16x16x4_f32(false, a, false, b,
                                                  (short)0, c, false, false);
    }

    float bs = bias[col0 + mn];
    #pragma unroll
    for (int r = 0; r < 8; ++r) {
        int row = row0 + r + half * 8;     // D: VGPR r -> M=r (lanes 0-15), M=8+r (lanes 16-31)
        float v = c[r] + bs;
        if (do_relu) v = fmaxf(v, 0.0f);
        out[(long)row * HD + col0 + mn] = v;
    }
}

// ---------------------------------------------------------------- batch norm
__global__ void bn_stats(const float* __restrict__ x, float* __restrict__ stats) {
    int c = threadIdx.x;                    // 128 threads, channel-owning
    int rbase = blockIdx.x * 128;
    int rend  = min(rbase + 128, NN);
    float s = 0.0f, sq = 0.0f;
    for (int r = rbase; r < rend; ++r) {
        float v = x[(long)r * HD + c];      // coalesced: lanes read consecutive addrs
        s += v; sq += v * v;
    }
    atomicAdd(&stats[c], s);
    atomicAdd(&stats[HD + c], sq);
}

__global__ void bn_finalize(const float* __restrict__ stats, const float* __restrict__ g,
                            const float* __restrict__ b, float* __restrict__ ss) {
    int c = threadIdx.x;  // 128
    float mean = stats[c] * (1.0f / (float)NN);
    float var  = stats[HD + c] * (1.0f / (float)NN) - mean * mean;
    float sc   = g[c] * rsqrtf(var + 1e-5f);
    ss[c]      = sc;
    ss[HD + c] = b[c] - mean * sc;
}

__global__ void bn_apply_relu(float* __restrict__ x, const float* __restrict__ ss) {
    long i = (long)blockIdx.x * blockDim.x + threadIdx.x;
    if (i >= (long)NH) return;
    int c = (int)(i & (HD - 1));
    float v = x[i] * ss[c] + ss[HD + c];
    x[i] = fmaxf(v, 0.0f);
}

// ---------------------------------------------------------------- final pool + MLP
// mean(agg(h)) over nodes == (1/N) * sum_v outdeg(v) * h[v] -> degree histogram trick
__global__ void deg_count(const int* __restrict__ src, int* __restrict__ deg) {
    int e = blockIdx.x * blockDim.x + threadIdx.x;
    if (e < EE) atomicAdd(&deg[src[e]], 1);
}

__global__ void colsum(const float* __restrict__ h, const int* __restrict__ deg,
                       float* __restrict__ sums) {
    int c = threadIdx.x;  // 128
    int rbase = blockIdx.x * 128;
    int rend  = min(rbase + 128, NN);
    float s0 = 0.0f, s1 = 0.0f;
    for (int r = rbase; r < rend; ++r) {
        float v = h[(long)r * HD + c];
        s0 += v;
        s1 += (float)deg[r] * v;
    }
    atomicAdd(&sums[c], s0);
    atomicAdd(&sums[HD + c], s1);
}

__global__ void final_mlp(const float* __restrict__ sums,
                          const float* __restrict__ W1, const float* __restrict__ b1,
                          const float* __restrict__ W2, const float* __restrict__ b2,
                          const float* __restrict__ W3, const float* __restrict__ b3,
                          float* __restrict__ out) {
    __shared__ float g[256];
    __shared__ float t1[128];
    __shared__ float t2[128];
    __shared__ float red[128];
    int t = threadIdx.x;  // 128
    g[t]       = sums[t]       * (1.0f / (float)NN);
    g[128 + t] = sums[128 + t] * (1.0f / (float)NN);
    __syncthreads();
    float acc = b1[t];
    for (int k = 0; k < 256; ++k) acc += g[k] * W1[k * HD + t];
    t1[t] = fmaxf(acc, 0.0f);
    __syncthreads();
    acc = b2[t];
    for (int k = 0; k < 128; ++k) acc += t1[k] * W2[k * HD + t];
    t2[t] = fmaxf(acc, 0.0f);
    __syncthreads();
    red[t] = t2[t] * W3[t];
    __syncthreads();
    for (int s = 64; s > 0; s >>= 1) {
        if (t < s) red[t] += red[t + s];
        __syncthreads();
    }
    if (t == 0) out[0] = red[0] + b3[0];
}

// ---------------------------------------------------------------- host driver
extern "C" void kernel_launch(void* const* d_in, const int* in_sizes, int n_in,
                              void* d_out, int out_size, void* d_ws, size_t ws_size,
                              hipStream_t stream) {
    (void)in_sizes; (void)n_in; (void)out_size; (void)ws_size;
    const float* x  = (const float*)d_in[0];
    const int* ei   = (const int*)d_in[1];
    const int* src  = ei;
    const int* dst  = ei + EE;
    auto F = [&](int i) { return (const float*)d_in[i]; };

    float* ws     = (float*)d_ws;
    float* bufA   = ws;
    float* bufAgg = ws + (size_t)NH;
    float* bufB   = ws + 2 * (size_t)NH;
    float* stats  = ws + 3 * (size_t)NH;
    float* ss     = stats + 256;
    float* sums   = ss + 256;
    int*   deg    = (int*)(sums + 256);

    const int ZB   = (NH + 255) / 256;      // elementwise grid
    const int SB   = (EE * 32) / 256;       // scatter grid: 200000 blocks
    const int GB   = NN / 16;               // gemm grid: 6250 strips (exact)
    const int RB   = (NN + 127) / 128;      // column-reduction grid

    auto run_bn = [&](float* buf, int gidx) {
        zero_f32<<<1, 256, 0, stream>>>(stats, 256);
        bn_stats<<<RB, 128, 0, stream>>>(buf, stats);
        bn_finalize<<<1, 128, 0, stream>>>(stats, F(gidx), F(gidx + 1), ss);
        bn_apply_relu<<<ZB, 256, 0, stream>>>(buf, ss);
    };
    auto run_mp = [&](const float* cur, float* nxt, int widx, int bnidx) {
        zero_f32<<<ZB, 256, 0, stream>>>(bufAgg, NH);
        scatter_add<<<SB, 256, 0, stream>>>(cur, src, dst, bufAgg);
        gemm_mp<<<GB, 256, 0, stream>>>(cur, bufAgg, F(widx), F(widx + 1), nxt, 256, 0);
        run_bn(nxt, bnidx);
    };

    // GraphTransfer trunk
    run_mp(x,    bufA, 2, 4);     // unite      -> bn1
    run_mp(bufA, bufB, 6, 8);     // graphconv1 -> bn2
    run_mp(bufB, bufA, 10, 12);   // graphconv2 -> bn3
    // SubGraph block
    run_mp(bufA, bufB, 14, 16);   // s_unite    -> sbn1
    run_mp(bufB, bufA, 18, 20);   // s_gconv    -> sbn2
    run_mp(bufA, bufB, 22, 20);   // s_transfer -> sbn2 (shared params, fresh stats)
    // linear (+relu), then bn(sbn3)+relu
    gemm_mp<<<GB, 256, 0, stream>>>(bufB, nullptr, F(24), F(25), bufA, 128, 1);
    run_bn(bufA, 26);

    // linmaps0 mean pool via outdegree-weighted colsum, then 3-layer MLP
    zero_f32<<<(NN + 255) / 256, 256, 0, stream>>>((float*)deg, NN);
    zero_f32<<<1, 256, 0, stream>>>(sums, 256);
    deg_count<<<(EE + 255) / 256, 256, 0, stream>>>(src, deg);
    colsum<<<RB, 128, 0, stream>>>(bufA, deg, sums);
    final_mlp<<<1, 128, 0, stream>>>(sums, F(28), F(29), F(30), F(31), F(32), F(33),
                                     (float*)d_out);
}